// VectorQuantizer_83708912599848
// MI455X (gfx1250) — compile-verified
//
#include <hip/hip_runtime.h>
#include <hip/hip_bf16.h>

typedef float v2f __attribute__((ext_vector_type(2)));
typedef float v8f __attribute__((ext_vector_type(8)));

#define N_E    1024
#define E_DIM  64
#define N_TOK  32768
#define SK_ITERS 100
#define LOG2K  10.0f   // log2(1024)
#define LOG2B  15.0f   // log2(32768)

// workspace byte offsets
#define OFF_D      ((size_t)0)                       // 1024*32768 f32 = 134217728 B
#define OFF_F      ((size_t)134217728)               // 1024 f32
#define OFF_G      (OFF_F + 4096)                    // 32768 f32
#define OFF_XN     (OFF_G + 131072)                  // 32768 f32
#define OFF_EN     (OFF_XN + 131072)                 // 1024 f32
#define OFF_PAR    (OFF_EN + 4096)                   // [0]=c02 [1]=beta2 (f32), [2]=gmin [3]=gmax (u32)
#define OFF_USAGE  (OFF_PAR + 16)                    // 1024 int
#define OFF_LOSS   (OFF_USAGE + 4096)                // 1 f32 (+pad)
#define OFF_IND    (OFF_LOSS + 16)                   // 32768 int

// ---------------- init ----------------
__global__ void __launch_bounds__(256) vq_init(float* g, float* f, int* usage,
                                               float* loss, unsigned* mm) {
    int t = blockIdx.x * 256 + threadIdx.x;
    if (t < N_TOK) g[t] = 0.0f;
    if (t < N_E) { f[t] = 0.0f; usage[t] = 0; }
    if (t == 0) { loss[0] = 0.0f; mm[0] = 0x7F7FFFFFu; mm[1] = 0u; }
}

// ---------------- row norms ----------------
__global__ void __launch_bounds__(256) vq_norms(const float* __restrict__ x,
                                                const float* __restrict__ emb,
                                                float* __restrict__ xn,
                                                float* __restrict__ en) {
    int t = blockIdx.x * 256 + threadIdx.x;
    if (t < N_TOK) {
        const float4* p = (const float4*)(x + (size_t)t * E_DIM);
        float s = 0.0f;
#pragma unroll
        for (int i = 0; i < 16; ++i) {
            float4 v = p[i];
            s += v.x * v.x + v.y * v.y + v.z * v.z + v.w * v.w;
        }
        xn[t] = s;
    } else {
        int k = t - N_TOK;
        if (k < N_E) {
            const float4* p = (const float4*)(emb + (size_t)k * E_DIM);
            float s = 0.0f;
#pragma unroll
            for (int i = 0; i < 16; ++i) {
                float4 v = p[i];
                s += v.x * v.x + v.y * v.y + v.z * v.z + v.w * v.w;
            }
            en[k] = s;
        }
    }
}

// ---------------- GEMM (fp32 WMMA) + distance + global min/max ----------------
// dmat[k][t] = ||x_t||^2 + ||e_k||^2 - 2 * <x_t, e_k>, layout [N_E][N_TOK]
__global__ void __launch_bounds__(256) vq_gemm(const float* __restrict__ x,
                                               const float* __restrict__ emb,
                                               const float* __restrict__ xn,
                                               const float* __restrict__ en,
                                               float* __restrict__ dmat,
                                               unsigned* mm) {
    const int lane = threadIdx.x & 31;
    const int wv   = threadIdx.x >> 5;
    const int bm   = blockIdx.x >> 8;                    // code tile 0..63
    const int bn   = ((blockIdx.x & 255) << 3) | wv;     // token tile 0..2047
    const int M0 = bm << 4, N0 = bn << 4;
    const int r  = lane & 15;       // M (codes) for A, N (tokens) for B, col for C/D
    const int hi = lane >> 4;       // K-half selector / row-half for C/D

    const float* ap = emb + (size_t)(M0 + r) * E_DIM + 2 * hi;  // A: 16x4 tile rows
    const float* bp = x   + (size_t)(N0 + r) * E_DIM + 2 * hi;  // B: 4x16 tile cols

    v8f acc = {0.f, 0.f, 0.f, 0.f, 0.f, 0.f, 0.f, 0.f};
#pragma unroll
    for (int kk = 0; kk < E_DIM; kk += 4) {
        v2f a = *(const v2f*)(ap + kk);
        v2f b = *(const v2f*)(bp + kk);
        acc = __builtin_amdgcn_wmma_f32_16x16x4_f32(
            false, a, false, b, (short)0, acc, false, false);
    }

    const float xnt = xn[N0 + r];
    float lmin = 3.4e38f, lmax = -3.4e38f;
#pragma unroll
    for (int v = 0; v < 8; ++v) {
        int k = M0 + v + 8 * hi;                 // C/D layout: M = v + 8*(lane>>4)
        float dv = xnt + en[k] - 2.0f * acc[v];
        dmat[(size_t)k * N_TOK + N0 + r] = dv;
        lmin = fminf(lmin, dv);
        lmax = fmaxf(lmax, dv);
    }

    __shared__ float smin[256], smax[256];
    smin[threadIdx.x] = lmin; smax[threadIdx.x] = lmax;
    __syncthreads();
    for (int off = 128; off; off >>= 1) {
        if (threadIdx.x < off) {
            smin[threadIdx.x] = fminf(smin[threadIdx.x], smin[threadIdx.x + off]);
            smax[threadIdx.x] = fmaxf(smax[threadIdx.x], smax[threadIdx.x + off]);
        }
        __syncthreads();
    }
    if (threadIdx.x == 0) {
        atomicMin(&mm[0], __float_as_uint(smin[0]));   // d >= 0: uint order == float order
        atomicMax(&mm[1], __float_as_uint(smax[0]));
    }
}

// ---------------- centering constants ----------------
__global__ void vq_params(const unsigned* mm, float* par) {
    float mn = __uint_as_float(mm[0]);
    float mx = __uint_as_float(mm[1]);
    float middle = 0.5f * (mx + mn);
    float amplitude = mx - middle + 1e-5f;
    // exp argument (base 2): arg = (middle - d)/(amplitude*eps) * log2(e)
    float beta2 = 1.44269504088896340736f / (amplitude * 0.01f);
    par[0] = middle * beta2;   // c02
    par[1] = beta2;
}

__device__ __forceinline__ void lse_update(float& m, float& s, float v) {
    float nm = fmaxf(m, v);
    s = s * __builtin_exp2f(m - nm) + __builtin_exp2f(v - nm);
    m = nm;
}
__device__ __forceinline__ void lse_merge(float& m, float& s, float m2, float s2) {
    float nm = fmaxf(m, m2);
    s = s * __builtin_exp2f(m - nm) + s2 * __builtin_exp2f(m2 - nm);
    m = nm;
}

// ---------------- Sinkhorn: per-code (row) update ----------------
// f[k] = -log2( sum_t 2^(c0 - beta*d[k][t] + g[t]) ) - LOG2K
__global__ void __launch_bounds__(256) vq_row(const float* __restrict__ dmat,
                                              const float* __restrict__ g,
                                              float* __restrict__ f,
                                              const float* __restrict__ par) {
    const int k = blockIdx.x;
    const float c0 = par[0], beta = par[1];
    const float* row = dmat + (size_t)k * N_TOK;
    float m0 = -__builtin_inff(), s0 = 0.f;
    float m1 = -__builtin_inff(), s1 = 0.f;
    for (int t = threadIdx.x; t < N_TOK; t += 512) {
        float v0 = c0 - beta * row[t]       + g[t];
        float v1 = c0 - beta * row[t + 256] + g[t + 256];
        lse_update(m0, s0, v0);
        lse_update(m1, s1, v1);
    }
    lse_merge(m0, s0, m1, s1);
    __shared__ float sm[256], ss[256];
    sm[threadIdx.x] = m0; ss[threadIdx.x] = s0;
    __syncthreads();
    for (int off = 128; off; off >>= 1) {
        if (threadIdx.x < off) {
            float m = sm[threadIdx.x], s = ss[threadIdx.x];
            lse_merge(m, s, sm[threadIdx.x + off], ss[threadIdx.x + off]);
            sm[threadIdx.x] = m; ss[threadIdx.x] = s;
        }
        __syncthreads();
    }
    if (threadIdx.x == 0) f[k] = -(sm[0] + __builtin_log2f(ss[0])) - LOG2K;
}

// ---------------- Sinkhorn: per-token (col) update ----------------
// g[t] = -log2( sum_k 2^(c0 - beta*d[k][t] + f[k]) ) - LOG2B
__global__ void __launch_bounds__(256) vq_col(const float* __restrict__ dmat,
                                              const float* __restrict__ f,
                                              float* __restrict__ g,
                                              const float* __restrict__ par) {
    __shared__ float fs[N_E];
    for (int i = threadIdx.x; i < N_E; i += 256) fs[i] = f[i];
    __syncthreads();
    const int t = blockIdx.x * 256 + threadIdx.x;
    const float c0 = par[0], beta = par[1];
    const float* col = dmat + t;
    float m0 = -__builtin_inff(), s0 = 0.f;
    float m1 = -__builtin_inff(), s1 = 0.f;
    float m2 = -__builtin_inff(), s2 = 0.f;
    float m3 = -__builtin_inff(), s3 = 0.f;
    for (int k = 0; k < N_E; k += 4) {
        float v0 = c0 - beta * col[(size_t)(k + 0) * N_TOK] + fs[k + 0];
        float v1 = c0 - beta * col[(size_t)(k + 1) * N_TOK] + fs[k + 1];
        float v2 = c0 - beta * col[(size_t)(k + 2) * N_TOK] + fs[k + 2];
        float v3 = c0 - beta * col[(size_t)(k + 3) * N_TOK] + fs[k + 3];
        lse_update(m0, s0, v0);
        lse_update(m1, s1, v1);
        lse_update(m2, s2, v2);
        lse_update(m3, s3, v3);
    }
    lse_merge(m0, s0, m1, s1);
    lse_merge(m2, s2, m3, s3);
    lse_merge(m0, s0, m2, s2);
    g[t] = -(m0 + __builtin_log2f(s0)) - LOG2B;
}

// ---------------- argmax + usage counts ----------------
__global__ void __launch_bounds__(256) vq_argmax(const float* __restrict__ dmat,
                                                 const float* __restrict__ f,
                                                 int* __restrict__ ind,
                                                 float* __restrict__ out_ind,
                                                 int* __restrict__ usage,
                                                 const float* __restrict__ par) {
    __shared__ float fs[N_E];
    for (int i = threadIdx.x; i < N_E; i += 256) fs[i] = f[i];
    __syncthreads();
    const int t = blockIdx.x * 256 + threadIdx.x;
    const float beta = par[1];
    const float* col = dmat + t;
    float best = -__builtin_inff();
    int bi = 0;
    for (int k = 0; k < N_E; ++k) {
        float v = fs[k] - beta * col[(size_t)k * N_TOK];
        if (v > best) { best = v; bi = k; }
    }
    ind[t] = bi;
    out_ind[t] = (float)bi;
    atomicAdd(&usage[bi], 1);
}

// ---------------- gather x_q + loss accumulation ----------------
__global__ void __launch_bounds__(256) vq_gather(const float* __restrict__ x,
                                                 const float* __restrict__ emb,
                                                 const int* __restrict__ ind,
                                                 float* __restrict__ out_xq,
                                                 float* __restrict__ loss) {
    int i = blockIdx.x * 256 + threadIdx.x;       // 0 .. 2097151
    int t = i >> 6, dd = i & 63;
    int k = ind[t];
    float e  = emb[(size_t)k * E_DIM + dd];
    float xv = x[i];
    out_xq[i] = xv + (e - xv);                    // straight-through arithmetic
    float df = e - xv;
    __shared__ float sr[256];
    sr[threadIdx.x] = df * df;
    __syncthreads();
    for (int off = 128; off; off >>= 1) {
        if (threadIdx.x < off) sr[threadIdx.x] += sr[threadIdx.x + off];
        __syncthreads();
    }
    if (threadIdx.x == 0) atomicAdd(loss, sr[0]);
}

// ---------------- final outputs ----------------
__global__ void __launch_bounds__(256) vq_final(const int* __restrict__ usage_in,
                                                const int* __restrict__ usage_ws,
                                                const float* __restrict__ loss,
                                                float* __restrict__ out_usage,
                                                float* __restrict__ out_loss) {
    int k = blockIdx.x * 256 + threadIdx.x;
    if (k < N_E) out_usage[k] = (float)(usage_in[k] + usage_ws[k]);
    if (k == 0) out_loss[0] = 2.0f * loss[0] / 2097152.0f;   // codebook + mu*commitment
}

extern "C" void kernel_launch(void* const* d_in, const int* in_sizes, int n_in,
                              void* d_out, int out_size, void* d_ws, size_t ws_size,
                              hipStream_t stream) {
    const float* x        = (const float*)d_in[0];
    const float* emb      = (const float*)d_in[1];
    const int*   usage_in = (const int*)d_in[4];

    float* out       = (float*)d_out;
    float* out_xq    = out;                       // 2097152
    float* out_loss  = out + 2097152;             // 1
    float* out_ind   = out + 2097153;             // 32768
    float* out_usage = out + 2097153 + 32768;     // 1024

    char* ws = (char*)d_ws;
    float*    dmat    = (float*)(ws + OFF_D);
    float*    f       = (float*)(ws + OFF_F);
    float*    g       = (float*)(ws + OFF_G);
    float*    xn      = (float*)(ws + OFF_XN);
    float*    en      = (float*)(ws + OFF_EN);
    float*    par     = (float*)(ws + OFF_PAR);
    unsigned* mm      = (unsigned*)(ws + OFF_PAR) + 2;
    int*      usage   = (int*)(ws + OFF_USAGE);
    float*    lossacc = (float*)(ws + OFF_LOSS);
    int*      ind     = (int*)(ws + OFF_IND);

    vq_init <<<128, 256, 0, stream>>>(g, f, usage, lossacc, mm);
    vq_norms<<<132, 256, 0, stream>>>(x, emb, xn, en);
    vq_gemm <<<16384, 256, 0, stream>>>(x, emb, xn, en, dmat, mm);
    vq_params<<<1, 1, 0, stream>>>(mm, par);
    for (int it = 0; it < SK_ITERS; ++it) {
        vq_row<<<1024, 256, 0, stream>>>(dmat, g, f, par);
        vq_col<<<128, 256, 0, stream>>>(dmat, f, g, par);
    }
    vq_argmax<<<128, 256, 0, stream>>>(dmat, f, ind, out_ind, usage, par);
    vq_gather<<<8192, 256, 0, stream>>>(x, emb, ind, out_xq, lossacc);
    vq_final <<<4, 256, 0, stream>>>(usage_in, usage, lossacc, out_usage, out_loss);
}